// PointNetSetAbstraction_50809463111977
// MI455X (gfx1250) — compile-verified
//
#include <hip/hip_runtime.h>
#include <hip/hip_bf16.h>
#include <stdint.h>

#define B_   16
#define N_   4096
#define S_   512
#define K_   32
#define R2_  0.04f
#define EPS_ 1e-5f
#define M_   (B_*S_*K_)   /* 262144 rows through the MLP */

typedef float v2f __attribute__((ext_vector_type(2)));
typedef float v8f __attribute__((ext_vector_type(8)));

static __device__ __forceinline__ v8f wmma4(v2f a, v2f b, v8f c) {
  // V_WMMA_F32_16X16X4_F32: full fp32 matrix op (CDNA5)
  return __builtin_amdgcn_wmma_f32_16x16x4_f32(
      /*neg_a=*/false, a, /*neg_b=*/false, b,
      /*c_mod=*/(short)0, c, /*reuse_a=*/false, /*reuse_b=*/false);
}

// ---------------------------------------------------------------------------
// 1) Farthest point sampling: one workgroup per batch, points in LDS,
//    block-wide packed argmax each of the 512 sequential steps.
// ---------------------------------------------------------------------------
__global__ __launch_bounds__(1024) void fps_kernel(const float* __restrict__ xyz,
                                                   int* __restrict__ fps_idx) {
  const int b = blockIdx.x;
  const float* base = xyz + (size_t)b * 3 * N_;   // layout (3, N)
  __shared__ float sx[N_], sy[N_], sz[N_];
  __shared__ unsigned long long red[32];
  __shared__ int sfar;
  const int t = threadIdx.x;
  for (int i = t; i < N_; i += 1024) {
    sx[i] = base[i]; sy[i] = base[N_ + i]; sz[i] = base[2 * N_ + i];
  }
  float dist[4];
#pragma unroll
  for (int j = 0; j < 4; ++j) dist[j] = 1e10f;
  int far = 0;
  __syncthreads();
  for (int s = 0; s < S_; ++s) {
    if (t == 0) fps_idx[b * S_ + s] = far;
    const float px = sx[far], py = sy[far], pz = sz[far];
    unsigned long long best = 0ull;
#pragma unroll
    for (int j = 0; j < 4; ++j) {
      const int i = t + j * 1024;
      const float dx = sx[i] - px, dy = sy[i] - py, dz = sz[i] - pz;
      const float d = dx * dx + dy * dy + dz * dz;
      dist[j] = fminf(dist[j], d);
      // max dist, smallest index on ties (matches jnp.argmax)
      const unsigned long long key =
          ((unsigned long long)__float_as_uint(dist[j]) << 32) |
          (unsigned)(0xFFFFFFFFu - (unsigned)i);
      best = best > key ? best : key;
    }
#pragma unroll
    for (int off = 16; off > 0; off >>= 1) {
      const unsigned long long o = __shfl_xor(best, off, 32);
      best = best > o ? best : o;
    }
    if ((t & 31) == 0) red[t >> 5] = best;
    __syncthreads();
    if (t < 32) {
      unsigned long long v = red[t];
#pragma unroll
      for (int off = 16; off > 0; off >>= 1) {
        const unsigned long long o = __shfl_xor(v, off, 32);
        v = v > o ? v : o;
      }
      if (t == 0) sfar = (int)(0xFFFFFFFFu - (unsigned)(v & 0xFFFFFFFFu));
    }
    __syncthreads();
    far = sfar;
  }
}

// ---------------------------------------------------------------------------
// 2) Write new_xyz output (B,3,S) from fps indices.
// ---------------------------------------------------------------------------
__global__ __launch_bounds__(256) void newxyz_kernel(const float* __restrict__ xyz,
                                                     const int* __restrict__ fps_idx,
                                                     float* __restrict__ out0) {
  const int idx = blockIdx.x * 256 + threadIdx.x;
  if (idx >= B_ * S_) return;
  const int b = idx / S_, s = idx - b * S_;
  const int i = fps_idx[idx];
  const float* xb = xyz + (size_t)b * 3 * N_;
  float* ob = out0 + (size_t)b * 3 * S_;
  ob[s] = xb[i];
  ob[S_ + s] = xb[N_ + i];
  ob[2 * S_ + s] = xb[2 * N_ + i];
}

// ---------------------------------------------------------------------------
// 3) Ball query + grouping: one wave per (b,s). Ballot/prefix-popcount keeps
//    the first K in-radius indices in ascending index order (== sorted take-K).
//    feat layout: M x 8 (cols 6..7 zero-padded for the K=4 WMMA chunks).
// ---------------------------------------------------------------------------
__global__ __launch_bounds__(256) void group_kernel(const float* __restrict__ xyz,
                                                    const float* __restrict__ pts,
                                                    const int* __restrict__ fps_idx,
                                                    float* __restrict__ feat) {
  const int gw = (blockIdx.x * 256 + threadIdx.x) >> 5;   // (b,s) pair id
  const int lane = threadIdx.x & 31;
  const int b = gw / S_, s = gw - b * S_;
  const float* xb = xyz + (size_t)b * 3 * N_;
  const float* pb = pts + (size_t)b * 3 * N_;
  const int ci = fps_idx[b * S_ + s];
  const float qx = xb[ci], qy = xb[N_ + ci], qz = xb[2 * N_ + ci];
  const float a2 = qx * qx + qy * qy + qz * qz;
  __shared__ int sidx[8][K_];
  int* myidx = sidx[threadIdx.x >> 5];
  int cnt = 0;
  for (int n0 = 0; n0 < N_ && cnt < K_; n0 += 32) {
    const int i = n0 + lane;
    const float x = xb[i], y = xb[N_ + i], z = xb[2 * N_ + i];
    const float b2 = x * x + y * y + z * z;
    const float dot = qx * x + qy * y + qz * z;
    const float sq = a2 + b2 - 2.0f * dot;         // same formula as reference
    const bool hit = !(sq > R2_);
    const unsigned mask = (unsigned)__ballot(hit); // wave32 -> low 32 bits
    const int pos = cnt + __popc(mask & ((1u << lane) - 1u));
    if (hit && pos < K_) myidx[pos] = i;
    cnt += __popc(mask);
  }
  __syncthreads();
  const int first = myidx[0];                       // centroid always within r
  const int nidx = (lane < cnt) ? myidx[lane] : first;
  float* f = feat + ((size_t)gw * K_ + lane) * 8;
  f[0] = xb[nidx] - qx;
  f[1] = xb[N_ + nidx] - qy;
  f[2] = xb[2 * N_ + nidx] - qz;
  f[3] = pb[nidx];
  f[4] = pb[N_ + nidx];
  f[5] = pb[2 * N_ + nidx];
  f[6] = 0.f;
  f[7] = 0.f;
}

// ---------------------------------------------------------------------------
// WMMA fp32 GEMM: D[m,o] = sum_c X[m,c]*W[o,c] + bias[o].
// One wave per 16x16 (M x O) tile; K in chunks of 4 per the fp32 A layout:
//   lane half (lane>=16) selects K pair {2h, 2h+1}; row = lane & 15.
// ---------------------------------------------------------------------------
template <int CIN, int COUT>
__global__ __launch_bounds__(256) void gemm_store_kernel(const float* __restrict__ X,
                                                         const float* __restrict__ W,
                                                         const float* __restrict__ bias,
                                                         float* __restrict__ H) {
  const int wid = (blockIdx.x * 256 + threadIdx.x) >> 5;
  const int lane = threadIdx.x & 31;
  const int otiles = COUT >> 4;
  const int mt = wid / otiles;
  const int ot = wid - mt * otiles;
  const int m0 = mt << 4, o0 = ot << 4;
  const int kh = lane >> 4;
  const int lrow = lane & 15;
  const float* xrow = X + (size_t)(m0 + lrow) * CIN;
  const float* wrow = W + (size_t)(o0 + lrow) * CIN;
  v8f acc = {};
#pragma unroll
  for (int k = 0; k < CIN; k += 4) {
    v2f a, bf;
    a.x = xrow[k + 2 * kh];
    a.y = xrow[k + 2 * kh + 1];
    bf.x = wrow[k + 2 * kh];
    bf.y = wrow[k + 2 * kh + 1];
    acc = wmma4(a, bf, acc);
  }
  const float bo = bias[o0 + lrow];
  float* hp = H + (size_t)(m0 + kh * 8) * COUT + o0 + lrow;
#pragma unroll
  for (int i = 0; i < 8; ++i) hp[(size_t)i * COUT] = acc[i] + bo;
}

// Layer-2 pass 1: same GEMM but only accumulate per-channel sum/sumsq
// (block shares one o-tile -> LDS pre-reduction, then 32 global atomics).
template <int CIN, int COUT>
__global__ __launch_bounds__(256) void gemm_stats_kernel(const float* __restrict__ X,
                                                         const float* __restrict__ W,
                                                         const float* __restrict__ bias,
                                                         float* __restrict__ stats) {
  const int otiles = COUT >> 4;
  const int ot = blockIdx.x % otiles;
  const int mt = (blockIdx.x / otiles) * 8 + (threadIdx.x >> 5);
  const int lane = threadIdx.x & 31;
  const int m0 = mt << 4, o0 = ot << 4;
  const int kh = lane >> 4, lrow = lane & 15;
  __shared__ float ssum[16], ssq[16];
  if (threadIdx.x < 16) { ssum[threadIdx.x] = 0.f; ssq[threadIdx.x] = 0.f; }
  __syncthreads();
  const float* xrow = X + (size_t)(m0 + lrow) * CIN;
  const float* wrow = W + (size_t)(o0 + lrow) * CIN;
  v8f acc = {};
#pragma unroll
  for (int k = 0; k < CIN; k += 4) {
    v2f a, bf;
    a.x = xrow[k + 2 * kh];
    a.y = xrow[k + 2 * kh + 1];
    bf.x = wrow[k + 2 * kh];
    bf.y = wrow[k + 2 * kh + 1];
    acc = wmma4(a, bf, acc);
  }
  const float bo = bias[o0 + lrow];
  float ls = 0.f, lq = 0.f;
#pragma unroll
  for (int i = 0; i < 8; ++i) { const float v = acc[i] + bo; ls += v; lq += v * v; }
  atomicAdd(&ssum[lrow], ls);
  atomicAdd(&ssq[lrow], lq);
  __syncthreads();
  if (threadIdx.x < 16) {
    atomicAdd(&stats[o0 + threadIdx.x], ssum[threadIdx.x]);
    atomicAdd(&stats[COUT + o0 + threadIdx.x], ssq[threadIdx.x]);
  }
}

// Layer-2 pass 2: GEMM fused with BN + ReLU + max over K + transposed store.
// Rows m0..m0+15 all belong to one (b,s) group (16 | K=32); reduce across
// lane halves with shfl_xor(16), combine the two K-halves via atomicMax on
// non-negative float bits (output region pre-zeroed).
template <int CIN, int COUT>
__global__ __launch_bounds__(256) void gemm_bnmax_kernel(const float* __restrict__ X,
                                                         const float* __restrict__ W,
                                                         const float* __restrict__ bias,
                                                         const float* __restrict__ stats,
                                                         const float* __restrict__ g,
                                                         const float* __restrict__ be,
                                                         float* __restrict__ outp) {
  const int otiles = COUT >> 4;
  const int ot = blockIdx.x % otiles;
  const int mt = (blockIdx.x / otiles) * 8 + (threadIdx.x >> 5);
  const int lane = threadIdx.x & 31;
  const int m0 = mt << 4, o0 = ot << 4;
  const int kh = lane >> 4, lrow = lane & 15;
  const float* xrow = X + (size_t)(m0 + lrow) * CIN;
  const float* wrow = W + (size_t)(o0 + lrow) * CIN;
  v8f acc = {};
#pragma unroll
  for (int k = 0; k < CIN; k += 4) {
    v2f a, bf;
    a.x = xrow[k + 2 * kh];
    a.y = xrow[k + 2 * kh + 1];
    bf.x = wrow[k + 2 * kh];
    bf.y = wrow[k + 2 * kh + 1];
    acc = wmma4(a, bf, acc);
  }
  const int o = o0 + lrow;
  const float bo = bias[o];
  const float mu = stats[o], inv = stats[COUT + o], gg = g[o], bb = be[o];
  float lm = 0.f;
#pragma unroll
  for (int i = 0; i < 8; ++i) {
    float v = acc[i] + bo;
    v = fmaxf(gg * (v - mu) * inv + bb, 0.f);
    lm = fmaxf(lm, v);
  }
  lm = fmaxf(lm, __shfl_xor(lm, 16, 32));
  if (kh == 0) {
    const int bs = m0 / K_;
    const int b = bs / S_, s = bs - b * S_;
    atomicMax((int*)(outp + (size_t)b * COUT * S_ + (size_t)o * S_ + s),
              __float_as_int(lm));
  }
}

// ---------------------------------------------------------------------------
// Per-channel sum/sumsq over stored activations (layers 0/1).
// 256 % C == 0 so each thread owns a fixed channel across the grid stride.
// ---------------------------------------------------------------------------
template <int C>
__global__ __launch_bounds__(256) void stats_kernel(const float* __restrict__ H,
                                                    float* __restrict__ stats) {
  __shared__ float ssum[C], ssq[C];
  const int t = threadIdx.x;
  for (int i = t; i < C; i += 256) { ssum[i] = 0.f; ssq[i] = 0.f; }
  __syncthreads();
  const int c = t % C;
  float ls = 0.f, lq = 0.f;
  const size_t total = (size_t)M_ * C;
  for (size_t e = (size_t)blockIdx.x * 256 + t; e < total;
       e += (size_t)gridDim.x * 256) {
    const float v = H[e];
    ls += v; lq += v * v;
  }
  atomicAdd(&ssum[c], ls);
  atomicAdd(&ssq[c], lq);
  __syncthreads();
  for (int i = t; i < C; i += 256) {
    atomicAdd(&stats[i], ssum[i]);
    atomicAdd(&stats[C + i], ssq[i]);
  }
}

__global__ void finalize_stats_kernel(float* stats, int C, float invM) {
  const int t = threadIdx.x;
  if (t < C) {
    const float mu = stats[t] * invM;
    const float var = stats[C + t] * invM - mu * mu;
    stats[t] = mu;
    stats[C + t] = rsqrtf(var + EPS_);
  }
}

template <int C>
__global__ __launch_bounds__(256) void bnrelu_kernel(float* __restrict__ H,
                                                     const float* __restrict__ stats,
                                                     const float* __restrict__ g,
                                                     const float* __restrict__ be) {
  const int t = threadIdx.x;
  const int c = t % C;
  const float mu = stats[c], inv = stats[C + c], gg = g[c], bb = be[c];
  const size_t total = (size_t)M_ * C;
  for (size_t e = (size_t)blockIdx.x * 256 + t; e < total;
       e += (size_t)gridDim.x * 256) {
    const float v = gg * (H[e] - mu) * inv + bb;
    H[e] = fmaxf(v, 0.f);
  }
}

__global__ void zero_kernel(float* __restrict__ p, int n) {
  const int i = blockIdx.x * 256 + threadIdx.x;
  if (i < n) p[i] = 0.f;
}

// ---------------------------------------------------------------------------
extern "C" void kernel_launch(void* const* d_in, const int* in_sizes, int n_in,
                              void* d_out, int out_size, void* d_ws, size_t ws_size,
                              hipStream_t stream) {
  (void)in_sizes; (void)n_in; (void)out_size; (void)ws_size;
  const float* xyz = (const float*)d_in[0];
  const float* pts = (const float*)d_in[1];
  const float* W0 = (const float*)d_in[2];
  const float* b0 = (const float*)d_in[3];
  const float* g0 = (const float*)d_in[4];
  const float* be0 = (const float*)d_in[5];
  const float* W1 = (const float*)d_in[6];
  const float* b1 = (const float*)d_in[7];
  const float* g1 = (const float*)d_in[8];
  const float* be1 = (const float*)d_in[9];
  const float* W2 = (const float*)d_in[10];
  const float* b2 = (const float*)d_in[11];
  const float* g2 = (const float*)d_in[12];
  const float* be2 = (const float*)d_in[13];

  float* out_xyz = (float*)d_out;              // (B,3,S)
  float* out_pts = out_xyz + (size_t)B_ * 3 * S_;  // (B,128,S)

  // workspace carve (~143 MB)
  char* w = (char*)d_ws;
  int* fps_idx = (int*)w;          w += (size_t)B_ * S_ * sizeof(int);     // 32 KB
  float* stats = (float*)w;        w += 256 * sizeof(float);               // 1 KB
  w = (char*)(((uintptr_t)w + 255) & ~(uintptr_t)255);
  float* feat = (float*)w;         w += (size_t)M_ * 8 * sizeof(float);    // 8 MB
  float* h0 = (float*)w;           w += (size_t)M_ * 64 * sizeof(float);   // 64 MB
  float* h1 = (float*)w;                                                   // 64 MB

  const float invM = 1.0f / (float)M_;

  // 1) FPS + new_xyz output
  fps_kernel<<<B_, 1024, 0, stream>>>(xyz, fps_idx);
  newxyz_kernel<<<(B_ * S_ + 255) / 256, 256, 0, stream>>>(xyz, fps_idx, out_xyz);

  // 2) ball query + grouping -> feat (M x 8)
  group_kernel<<<(B_ * S_) / 8, 256, 0, stream>>>(xyz, pts, fps_idx, feat);

  // 3) layer 0: (M,8) x (8,64)
  gemm_store_kernel<8, 64><<<(M_ / 16) * 4 / 8, 256, 0, stream>>>(feat, W0, b0, h0);
  zero_kernel<<<1, 256, 0, stream>>>(stats, 256);
  stats_kernel<64><<<256, 256, 0, stream>>>(h0, stats);
  finalize_stats_kernel<<<1, 128, 0, stream>>>(stats, 64, invM);
  bnrelu_kernel<64><<<2048, 256, 0, stream>>>(h0, stats, g0, be0);

  // 4) layer 1: (M,64) x (64,64)
  gemm_store_kernel<64, 64><<<(M_ / 16) * 4 / 8, 256, 0, stream>>>(h0, W1, b1, h1);
  zero_kernel<<<1, 256, 0, stream>>>(stats, 256);
  stats_kernel<64><<<256, 256, 0, stream>>>(h1, stats);
  finalize_stats_kernel<<<1, 128, 0, stream>>>(stats, 64, invM);
  bnrelu_kernel<64><<<2048, 256, 0, stream>>>(h1, stats, g1, be1);

  // 5) layer 2 two-pass: stats-only GEMM, then GEMM+BN+ReLU+maxK+transpose
  zero_kernel<<<1, 256, 0, stream>>>(stats, 256);
  gemm_stats_kernel<64, 128><<<(M_ / 16 / 8) * 8, 256, 0, stream>>>(h1, W2, b2, stats);
  finalize_stats_kernel<<<1, 128, 0, stream>>>(stats, 128, invM);
  zero_kernel<<<(B_ * 128 * S_ + 255) / 256, 256, 0, stream>>>(out_pts, B_ * 128 * S_);
  gemm_bnmax_kernel<64, 128><<<(M_ / 16 / 8) * 8, 256, 0, stream>>>(
      h1, W2, b2, stats, g2, be2, out_pts);
}